// FAGCN_28449863368914
// MI455X (gfx1250) — compile-verified
//
#include <hip/hip_runtime.h>
#include <math.h>

// ---------------------------------------------------------------------------
// FAGCN forward for MI455X (gfx1250, wave32).
//   raw  = relu(h @ t1_w^T + t1_b)               [N,128]  (fp32 WMMA GEMM)
//   loop 2 layers:
//     gd,gs = x @ gate_w[i][:128], x @ gate_w[i][128:]
//     z[dst] += x[src] * ((tanh(gd[dst]+gs[src]+b) + yn_term) * 0.5 * d[dst]*d[src])
//     x = EPS*raw + z
//   out = log_softmax(x @ t2_w^T + t2_b)          [N,64]
// ---------------------------------------------------------------------------

typedef __attribute__((ext_vector_type(2))) float v2f;
typedef __attribute__((ext_vector_type(8))) float v8f;

#define HIDN   128
#define INDIM  512
#define OUTD   64
#define EPSV   0.3f
#define TK     32        // K panel staged in LDS
#define LA_ST  36        // padded A row stride: banks 4r vs 4r+2 -> conflict free
#define LB_ST  288       // padded B pair-row stride: +288 dwords == +32 banks

// native f32 atomic add, no return (STOREcnt; drained by S_ENDPGM wait-idle)
__device__ __forceinline__ void atomic_add_f32(float* p, float v) {
    asm volatile("global_atomic_add_f32 %0, %1, off" :: "v"(p), "v"(v) : "memory");
}

// ---------------- Phase 1: x = relu(h @ t1_w^T + b) via V_WMMA_F32_16X16X4_F32
__global__ __launch_bounds__(256) void fagcn_gemm1(
    const float* __restrict__ h, const float* __restrict__ t1w,
    const float* __restrict__ t1b, float* __restrict__ xout, int N)
{
    __shared__ float lA[128 * LA_ST];     // [row][k], padded
    __shared__ float lBp[(TK / 2) * LB_ST]; // [k-pair][2n + (k&1)], padded

    const int tid  = threadIdx.x;
    const int wave = tid >> 5;            // 0..7  -> M tile within block
    const int lane = tid & 31;
    const int l16  = lane & 15;
    const int kh   = lane >> 4;           // half-wave: K offset 0 or 2
    const int mtile   = blockIdx.x * 8 + wave;
    const int rowBase = blockIdx.x * 128;

    v8f acc[8] = {};                      // 16x128 output per wave (8 N-tiles)

    const int r    = tid >> 1;            // cooperative-load row / col, 0..127
    const int half = (tid & 1) * 16;      // k sub-offset within panel

    for (int kp = 0; kp < INDIM; kp += TK) {
        // ---- stage A panel (h rows) and B panel (t1_w, k-pair interleaved)
        int gr = rowBase + r; if (gr > N - 1) gr = N - 1;
        const float* ap = h   + (size_t)gr * INDIM + kp + half;
        const float* bp = t1w + (size_t)r  * INDIM + kp + half;   // r = output col
#pragma unroll
        for (int j = 0; j < 16; j += 4) {
            float4 av = *(const float4*)(ap + j);
            float4 bv = *(const float4*)(bp + j);
            lA[r * LA_ST + half + j + 0] = av.x;
            lA[r * LA_ST + half + j + 1] = av.y;
            lA[r * LA_ST + half + j + 2] = av.z;
            lA[r * LA_ST + half + j + 3] = av.w;
            int p0 = (half + j) >> 1;     // even k -> pair index
            lBp[(p0 + 0) * LB_ST + 2 * r + 0] = bv.x;
            lBp[(p0 + 0) * LB_ST + 2 * r + 1] = bv.y;
            lBp[(p0 + 1) * LB_ST + 2 * r + 0] = bv.z;
            lBp[(p0 + 1) * LB_ST + 2 * r + 1] = bv.w;
        }
        __syncthreads();

        if (mtile * 16 < N) {             // wave-uniform -> EXEC stays all-1s
#pragma unroll
            for (int kk = 0; kk < TK; kk += 4) {
                // A frag (16x4 f32): v0 = K(2*kh), v1 = K(2*kh+1), M = l16
                v2f A = *(const v2f*)&lA[(wave * 16 + l16) * LA_ST + kk + 2 * kh];
                const int p = (kk >> 1) + kh;
#pragma unroll
                for (int t = 0; t < 8; ++t) {
                    // B frag: one aligned b64 -> even VGPR pair, no repacking
                    v2f B = *(const v2f*)&lBp[p * LB_ST + 2 * (t * 16 + l16)];
                    acc[t] = __builtin_amdgcn_wmma_f32_16x16x4_f32(
                        false, A, false, B, (short)0, acc[t], false, false);
                }
            }
        }
        __syncthreads();
    }

    // ---- epilogue: D layout VGPR ri -> M = ri + 8*kh, N = l16 (+16*t)
    if (mtile * 16 < N) {
#pragma unroll
        for (int t = 0; t < 8; ++t) {
            int col = t * 16 + l16;
            float bias = t1b[col];
#pragma unroll
            for (int ri = 0; ri < 8; ++ri) {
                int row = mtile * 16 + ri + 8 * kh;
                if (row < N) {
                    float v = acc[t][ri] + bias;
                    xout[(size_t)row * HIDN + col] = v > 0.f ? v : 0.f;
                }
            }
        }
    }
}

// ---------------- gate matvecs: gd[n] = x[n]·gw[:128], gs[n] = x[n]·gw[128:]
__global__ __launch_bounds__(256) void fagcn_gate(
    const float* __restrict__ x, const float* __restrict__ gw,
    float* __restrict__ gd, float* __restrict__ gs, int N)
{
    int gid  = blockIdx.x * 256 + threadIdx.x;
    int node = gid >> 5;
    int lane = threadIdx.x & 31;
    if (node >= N) return;

    float4 xv = *(const float4*)&x[(size_t)node * HIDN + lane * 4];
    float4 wd = *(const float4*)&gw[lane * 4];
    float4 ws = *(const float4*)&gw[HIDN + lane * 4];
    float a = xv.x * wd.x + xv.y * wd.y + xv.z * wd.z + xv.w * wd.w;
    float b = xv.x * ws.x + xv.y * ws.y + xv.z * ws.z + xv.w * ws.w;
#pragma unroll
    for (int m = 16; m > 0; m >>= 1) {
        a += __shfl_xor(a, m, 32);
        b += __shfl_xor(b, m, 32);
    }
    if (lane == 0) { gd[node] = a; gs[node] = b; }
}

// ---------------- edge scatter: z[dst] += x[src] * coef   (one wave per edge)
__global__ __launch_bounds__(256) void fagcn_edge(
    const float* __restrict__ x, const float* __restrict__ gd,
    const float* __restrict__ gs, const float* __restrict__ dn,
    const int* __restrict__ src, const int* __restrict__ dst,
    const int* __restrict__ yes_no,
    const float* __restrict__ yesw_p, const float* __restrict__ now_p,
    const float* __restrict__ gb_p, float* __restrict__ z, int E)
{
    int gid  = blockIdx.x * 256 + threadIdx.x;
    int e    = gid >> 5;
    int lane = threadIdx.x & 31;
    if (e >= E) return;

    int s = src[e], t = dst[e];
    float yn   = (float)yes_no[e];
    float ynt  = tanhf(yn * yesw_p[0] + (1.f - yn) * now_p[0]);
    float g    = tanhf(gd[t] + gs[s] + gb_p[0]);
    float coef = (g + ynt) * 0.5f * dn[t] * dn[s];

    float4 xv = *(const float4*)&x[(size_t)s * HIDN + lane * 4];
    float* zp = z + (size_t)t * HIDN + lane * 4;
    atomic_add_f32(zp + 0, xv.x * coef);
    atomic_add_f32(zp + 1, xv.y * coef);
    atomic_add_f32(zp + 2, xv.z * coef);
    atomic_add_f32(zp + 3, xv.w * coef);
}

// ---------------- x = EPS*raw + z
__global__ __launch_bounds__(256) void fagcn_combine(
    const float* __restrict__ raw, const float* __restrict__ z,
    float* __restrict__ xo, int total)
{
    int i = blockIdx.x * 256 + threadIdx.x;
    if (i < total) xo[i] = EPSV * raw[i] + z[i];
}

// ---------------- logits = x @ t2_w^T + b ; log_softmax rows of 64
__global__ __launch_bounds__(64) void fagcn_out(
    const float* __restrict__ x, const float* __restrict__ t2w,
    const float* __restrict__ t2b, float* __restrict__ out, int N)
{
    __shared__ float xs[HIDN];
    __shared__ float red[OUTD];
    int n = blockIdx.x;
    int t = threadIdx.x;                 // 0..63
    xs[t]      = x[(size_t)n * HIDN + t];
    xs[t + 64] = x[(size_t)n * HIDN + 64 + t];
    __syncthreads();

    float acc = t2b[t];
    const float* wr = t2w + (size_t)t * HIDN;
#pragma unroll 8
    for (int k = 0; k < HIDN; ++k) acc += xs[k] * wr[k];

    red[t] = acc; __syncthreads();
#pragma unroll
    for (int off = 32; off > 0; off >>= 1) {
        if (t < off) red[t] = fmaxf(red[t], red[t + off]);
        __syncthreads();
    }
    float mx = red[0]; __syncthreads();
    red[t] = expf(acc - mx); __syncthreads();
#pragma unroll
    for (int off = 32; off > 0; off >>= 1) {
        if (t < off) red[t] += red[t + off];
        __syncthreads();
    }
    float lse = mx + logf(red[0]);
    out[(size_t)n * OUTD + t] = acc - lse;
}

// ---------------------------------------------------------------------------
extern "C" void kernel_launch(void* const* d_in, const int* in_sizes, int n_in,
                              void* d_out, int out_size, void* d_ws, size_t ws_size,
                              hipStream_t stream)
{
    const float* h    = (const float*)d_in[0];
    const float* dn   = (const float*)d_in[1];
    const float* t1w  = (const float*)d_in[2];
    const float* t1b  = (const float*)d_in[3];
    const float* gw   = (const float*)d_in[4];   // [2, 256]
    const float* gb   = (const float*)d_in[5];   // [2]
    const float* t2w  = (const float*)d_in[6];
    const float* t2b  = (const float*)d_in[7];
    const float* yesw = (const float*)d_in[8];
    const float* now  = (const float*)d_in[9];
    const int*   src  = (const int*)d_in[10];
    const int*   dst  = (const int*)d_in[11];
    const int*   yn   = (const int*)d_in[12];
    const int N = in_sizes[1];     // d has N elements
    const int E = in_sizes[10];

    float* raw  = (float*)d_ws;                       // [N,128]
    float* xcur = raw  + (size_t)N * HIDN;            // [N,128]
    float* z    = xcur + (size_t)N * HIDN;            // [N,128]
    float* gdv  = z    + (size_t)N * HIDN;            // [N]
    float* gsv  = gdv  + N;                           // [N]

    fagcn_gemm1<<<(N + 127) / 128, 256, 0, stream>>>(h, t1w, t1b, raw, N);

    for (int i = 0; i < 2; ++i) {
        const float* xin = (i == 0) ? raw : xcur;
        fagcn_gate<<<(N + 7) / 8, 256, 0, stream>>>(xin, gw + i * 2 * HIDN, gdv, gsv, N);
        hipMemsetAsync(z, 0, (size_t)N * HIDN * sizeof(float), stream);
        fagcn_edge<<<(E + 7) / 8, 256, 0, stream>>>(xin, gdv, gsv, dn, src, dst, yn,
                                                    yesw, now, gb + i, z, E);
        int total = N * HIDN;
        fagcn_combine<<<(total + 255) / 256, 256, 0, stream>>>(raw, z, xcur, total);
    }

    fagcn_out<<<N, 64, 0, stream>>>(xcur, t2w, t2b, (float*)d_out, N);
}